// CGCNN_58815282151651
// MI455X (gfx1250) — compile-verified
//
#include <hip/hip_runtime.h>
#include <hip/hip_bf16.h>

// Problem constants (from reference)
#define N_NODES 40000
#define N_PAD   40064          // padded to multiple of 128 for WMMA tiling
#define E_EDGES 640000
#define G_GRAPH 256
#define H_DIM   128
#define L_CONV  5
#define FC_N    3
#define Z_DIM   257            // 2H+1

typedef __attribute__((ext_vector_type(2))) float v2f;
typedef __attribute__((ext_vector_type(8))) float v8f;
typedef __attribute__((ext_vector_type(4))) unsigned int u32x4;
typedef __attribute__((ext_vector_type(8))) int i32x8;
typedef __attribute__((ext_vector_type(4))) int i32x4;

#define LDS_PITCH 130          // 128 + 2 DWORD pad (TDM pad_amount=2) -> bank stride 2

// ---------------------------------------------------------------------------
// x[n,h] = embed[atoms[n], h]  (pad rows zeroed so WMMA tiles stay clean)
__global__ __launch_bounds__(256) void k_embed(const int* __restrict__ atoms,
                                               const float* __restrict__ embed,
                                               float* __restrict__ x)
{
    int idx = blockIdx.x * 256 + threadIdx.x;
    if (idx >= N_PAD * H_DIM) return;
    int n = idx >> 7, h = idx & 127;
    x[idx] = (n < N_NODES) ? embed[atoms[n] * H_DIM + h] : 0.0f;
}

// ea[e] = || pos[src] - pos[dst] ||
__global__ __launch_bounds__(256) void k_edge_attr(const float* __restrict__ pos,
                                                   const int* __restrict__ src,
                                                   const int* __restrict__ dst,
                                                   float* __restrict__ ea)
{
    int e = blockIdx.x * 256 + threadIdx.x;
    if (e >= E_EDGES) return;
    int s = src[e], d = dst[e];
    float dx = pos[s * 3 + 0] - pos[d * 3 + 0];
    float dy = pos[s * 3 + 1] - pos[d * 3 + 1];
    float dz = pos[s * 3 + 2] - pos[d * 3 + 2];
    ea[e] = sqrtf(dx * dx + dy * dy + dz * dz);
}

// ---------------------------------------------------------------------------
// Out[n,h] = sum_k X[n,k] * W[h*ldw + k]   (f32 WMMA 16x16x4)
// Block = 256 threads = 8 waves; wave w owns rows [blk*128 + w*16, +16), all 128 cols.
// Weight block (128x128, global row stride = ldw) DMA'd into LDS by the Tensor
// Data Mover with a 2-DWORD row pad (LDS pitch 130 -> conflict-free b64 reads).
__global__ __launch_bounds__(256) void k_gemm_wmma(const float* __restrict__ X,
                                                   const float* __restrict__ W,
                                                   int ldw,
                                                   float* __restrict__ Out)
{
    __shared__ float sW[128 * LDS_PITCH];

    if (threadIdx.x < 32) {  // one wave issues the TDM DMA (EXEC-independent op)
        unsigned long long wa = (unsigned long long)(const void*)W;
        unsigned int lds_off = (unsigned int)(unsigned long long)&sW[0];
        // D# group 0: count=1 | lds_addr | global_addr | type=2 ("image")
        u32x4 g0;
        g0.x = 1u;
        g0.y = lds_off;
        g0.z = (unsigned int)(wa & 0xFFFFFFFFull);
        g0.w = (unsigned int)((wa >> 32) & 0x01FFFFFFull) | (2u << 30);
        // D# group 1: data_size=4B, pad_enable, pad_interval=128 DW (code 6),
        //             pad_amount=2 DW (code 1); tensor 128x128, tile 128x128,
        //             tensor_dim0_stride = ldw
        i32x8 g1;
        g1[0] = (2 << 16) | (1 << 20) | (6 << 22) | (1 << 25);
        g1[1] = (128 << 16);   // [63:48]  tensor_dim0.lo16 = 128
        g1[2] = (128 << 16);   // [79:64]  dim0.hi=0 ; [95:80] tensor_dim1.lo16 = 128
        g1[3] = (128 << 16);   // [111:96] dim1.hi=0 ; [127:112] tile_dim0 = 128
        g1[4] = 128;           // [143:128] tile_dim1 = 128 ; tile_dim2 = 0
        g1[5] = ldw;           // [191:160] tensor_dim0_stride.lo32
        g1[6] = 0;             // stride.hi16 = 0 ; tensor_dim1_stride.lo16 = 0
        g1[7] = 0;
        i32x4 z4 = {0, 0, 0, 0};
        i32x8 z8 = {0, 0, 0, 0, 0, 0, 0, 0};
        __builtin_amdgcn_tensor_load_to_lds(g0, g1, z4, z4, z8, 0);
        __builtin_amdgcn_s_wait_tensorcnt(0);
    }
    __syncthreads();

    const int wave  = threadIdx.x >> 5;
    const int lane  = threadIdx.x & 31;
    const int m     = lane & 15;           // row within A tile / col within out tile
    const int khalf = (lane >> 4) << 1;    // K-pair selector per ISA 16x4 A-layout
    const int row0  = blockIdx.x * 128 + wave * 16;
    const float* xrow = X + (size_t)(row0 + m) * H_DIM + khalf;

    v8f acc[8] = {};
    for (int kk = 0; kk < 128; kk += 4) {
        v2f a = *(const v2f*)(xrow + kk);                 // global_load_b64
#pragma unroll
        for (int t = 0; t < 8; ++t) {
            int h = t * 16 + m;
            v2f b = *(const v2f*)(&sW[h * LDS_PITCH + kk + khalf]);  // ds_load_b64
            acc[t] = __builtin_amdgcn_wmma_f32_16x16x4_f32(
                false, a, false, b, (short)0, acc[t], false, false);
        }
    }
    // C/D layout: VGPR r -> (M = r + (lane<16 ? 0 : 8), N = lane & 15)
    const int mbase = (lane < 16) ? 0 : 8;
#pragma unroll
    for (int t = 0; t < 8; ++t)
#pragma unroll
        for (int r = 0; r < 8; ++r)
            Out[(size_t)(row0 + mbase + r) * H_DIM + t * 16 + m] = acc[t][r];
}

// ---------------------------------------------------------------------------
__global__ __launch_bounds__(256) void k_copy4(const float4* __restrict__ a,
                                               float4* __restrict__ b, int n4)
{
    int i = blockIdx.x * 256 + threadIdx.x;
    if (i < n4) b[i] = a[i];
}

__global__ __launch_bounds__(256) void k_zero(float* __restrict__ p, int n)
{
    int i = blockIdx.x * 256 + threadIdx.x;
    if (i < n) p[i] = 0.0f;
}

// ---------------------------------------------------------------------------
// One wave per edge; lane handles 4 channels (float4).
// pre_f = Pfd[dst] + Pfs[src] + ea*Wf[:,256] + bf ; gate = sigmoid(pre_f)
// pre_s likewise ; core = softplus(pre_s) ; atomic scatter gate*core into xnext[dst].
__global__ __launch_bounds__(256) void k_edge_msg(
    const float* __restrict__ Pfd, const float* __restrict__ Pfs,
    const float* __restrict__ Psd, const float* __restrict__ Pss,
    const float* __restrict__ ea,
    const int* __restrict__ src, const int* __restrict__ dst,
    const float* __restrict__ wf_col, const float* __restrict__ ws_col,
    const float* __restrict__ bf, const float* __restrict__ bs,
    float* __restrict__ xnext)
{
    __shared__ float s_wf[128], s_ws[128], s_bf[128], s_bs[128];
    if (threadIdx.x < 128) {
        s_wf[threadIdx.x] = wf_col[threadIdx.x * Z_DIM];
        s_ws[threadIdx.x] = ws_col[threadIdx.x * Z_DIM];
        s_bf[threadIdx.x] = bf[threadIdx.x];
        s_bs[threadIdx.x] = bs[threadIdx.x];
    }
    __syncthreads();

    int wave = threadIdx.x >> 5, lane = threadIdx.x & 31;
    int e = blockIdx.x * 8 + wave;
    if (e >= E_EDGES) return;
    int d = dst[e], s = src[e];
    float r = ea[e];
    int h = lane * 4;

    const float4 fd = *(const float4*)(Pfd + (size_t)d * H_DIM + h);
    const float4 fs = *(const float4*)(Pfs + (size_t)s * H_DIM + h);
    const float4 sd = *(const float4*)(Psd + (size_t)d * H_DIM + h);
    const float4 ss = *(const float4*)(Pss + (size_t)s * H_DIM + h);
    float af[4]  = {fd.x, fd.y, fd.z, fd.w};
    float bf4[4] = {fs.x, fs.y, fs.z, fs.w};
    float as[4]  = {sd.x, sd.y, sd.z, sd.w};
    float bs4[4] = {ss.x, ss.y, ss.z, ss.w};

    float* outp = xnext + (size_t)d * H_DIM + h;
#pragma unroll
    for (int j = 0; j < 4; ++j) {
        float zf = af[j] + bf4[j] + r * s_wf[h + j] + s_bf[h + j];
        float zs = as[j] + bs4[j] + r * s_ws[h + j] + s_bs[h + j];
        float gate = 1.0f / (1.0f + expf(-zf));
        float core = fmaxf(zs, 0.0f) + log1pf(expf(-fabsf(zs)));  // logaddexp(zs,0)
        atomicAdd(outp + j, gate * core);
    }
}

// ---------------------------------------------------------------------------
__global__ __launch_bounds__(256) void k_pool_sum(const float* __restrict__ x,
                                                  const int* __restrict__ batch,
                                                  float* __restrict__ gsum,
                                                  float* __restrict__ gcnt)
{
    int idx = blockIdx.x * 256 + threadIdx.x;
    if (idx >= N_NODES * H_DIM) return;
    int n = idx >> 7, h = idx & 127;
    int g = batch[n];
    atomicAdd(&gsum[(size_t)g * H_DIM + h], x[idx]);
    if (h == 0) atomicAdd(&gcnt[g], 1.0f);
}

__global__ __launch_bounds__(256) void k_pool_div(const float* __restrict__ gsum,
                                                  const float* __restrict__ gcnt,
                                                  float* __restrict__ g)
{
    int idx = blockIdx.x * 256 + threadIdx.x;
    if (idx >= G_GRAPH * H_DIM) return;
    g[idx] = gsum[idx] / fmaxf(gcnt[idx >> 7], 1.0f);
}

// out[h] = b[h] + sum_k in[g,k] * W[h,k]    (one block per graph)
__global__ __launch_bounds__(128) void k_fc(const float* __restrict__ in,
                                            const float* __restrict__ W,
                                            const float* __restrict__ b,
                                            float* __restrict__ out)
{
    __shared__ float sg[128];
    int g = blockIdx.x, h = threadIdx.x;
    sg[h] = in[(size_t)g * H_DIM + h];
    __syncthreads();
    float acc = b[h];
    for (int k = 0; k < H_DIM; ++k) acc = fmaf(sg[k], W[h * H_DIM + k], acc);
    out[(size_t)g * H_DIM + h] = acc;
}

__global__ __launch_bounds__(256) void k_out(const float* __restrict__ in,
                                             const float* __restrict__ ow,
                                             const float* __restrict__ ob,
                                             float* __restrict__ out)
{
    int g = threadIdx.x;
    if (g >= G_GRAPH) return;
    float acc = ob[0];
    for (int k = 0; k < H_DIM; ++k) acc = fmaf(in[(size_t)g * H_DIM + k], ow[k], acc);
    out[g] = acc;
}

// ---------------------------------------------------------------------------
extern "C" void kernel_launch(void* const* d_in, const int* in_sizes, int n_in,
                              void* d_out, int out_size, void* d_ws, size_t ws_size,
                              hipStream_t stream)
{
    const int*   atoms   = (const int*)d_in[0];
    const float* pos     = (const float*)d_in[1];
    const int*   ei      = (const int*)d_in[2];
    const int*   batch   = (const int*)d_in[3];
    const float* embed   = (const float*)d_in[4];
    const float* lin_f_w = (const float*)d_in[5];
    const float* lin_f_b = (const float*)d_in[6];
    const float* lin_s_w = (const float*)d_in[7];
    const float* lin_s_b = (const float*)d_in[8];
    const float* fc_w    = (const float*)d_in[9];
    const float* fc_b    = (const float*)d_in[10];
    const float* out_w   = (const float*)d_in[11];
    const float* out_b   = (const float*)d_in[12];
    float* out = (float*)d_out;

    // Workspace layout (floats). Total ~31.6M floats (~126 MB).
    const size_t NH = (size_t)N_PAD * H_DIM;
    float* ws   = (float*)d_ws;
    float* xA   = ws;                                  // NH
    float* xB   = xA + NH;                             // NH
    float* Pfd  = xB + NH;                             // NH
    float* Pfs  = Pfd + NH;                            // NH
    float* Psd  = Pfs + NH;                            // NH
    float* Pss  = Psd + NH;                            // NH
    float* ea   = Pss + NH;                            // E
    float* gsum = ea + E_EDGES;                        // G*H
    float* gcnt = gsum + (size_t)G_GRAPH * H_DIM;      // G   (zeroed with gsum)
    float* gA   = gcnt + G_GRAPH;                      // G*H
    float* gB   = gA + (size_t)G_GRAPH * H_DIM;        // G*H

    const int* srcI = ei;            // edge_index[0] = source j
    const int* dstI = ei + E_EDGES;  // edge_index[1] = target i

    k_embed<<<(N_PAD * H_DIM + 255) / 256, 256, 0, stream>>>(atoms, embed, xA);
    k_edge_attr<<<(E_EDGES + 255) / 256, 256, 0, stream>>>(pos, srcI, dstI, ea);

    float* xcur = xA;
    float* xnxt = xB;
    for (int l = 0; l < L_CONV; ++l) {
        const float* Wf  = lin_f_w + (size_t)l * H_DIM * Z_DIM;
        const float* Wsp = lin_s_w + (size_t)l * H_DIM * Z_DIM;
        const float* bfp = lin_f_b + (size_t)l * H_DIM;
        const float* bsp = lin_s_b + (size_t)l * H_DIM;

        // 4 node-level GEMMs replace the E x Z x H edge GEMM (421 -> 26 GFLOP).
        k_gemm_wmma<<<N_PAD / 128, 256, 0, stream>>>(xcur, Wf,        Z_DIM, Pfd);
        k_gemm_wmma<<<N_PAD / 128, 256, 0, stream>>>(xcur, Wf  + 128, Z_DIM, Pfs);
        k_gemm_wmma<<<N_PAD / 128, 256, 0, stream>>>(xcur, Wsp,       Z_DIM, Psd);
        k_gemm_wmma<<<N_PAD / 128, 256, 0, stream>>>(xcur, Wsp + 128, Z_DIM, Pss);

        k_copy4<<<(int)((NH / 4 + 255) / 256), 256, 0, stream>>>(
            (const float4*)xcur, (float4*)xnxt, (int)(NH / 4));
        k_edge_msg<<<E_EDGES / 8, 256, 0, stream>>>(Pfd, Pfs, Psd, Pss, ea, srcI, dstI,
                                                    Wf + 256, Wsp + 256, bfp, bsp, xnxt);
        float* t = xcur; xcur = xnxt; xnxt = t;
    }

    k_zero<<<((G_GRAPH * H_DIM + G_GRAPH) + 255) / 256, 256, 0, stream>>>(
        gsum, G_GRAPH * H_DIM + G_GRAPH);
    k_pool_sum<<<(N_NODES * H_DIM + 255) / 256, 256, 0, stream>>>(xcur, batch, gsum, gcnt);
    k_pool_div<<<(G_GRAPH * H_DIM + 255) / 256, 256, 0, stream>>>(gsum, gcnt, gA);

    float* gin = gA; float* gout = gB;
    for (int f = 0; f < FC_N; ++f) {
        k_fc<<<G_GRAPH, 128, 0, stream>>>(gin, fc_w + (size_t)f * H_DIM * H_DIM,
                                          fc_b + (size_t)f * H_DIM, gout);
        float* t = gin; gin = gout; gout = t;
    }
    k_out<<<1, 256, 0, stream>>>(gin, out_w, out_b, out);
}